// DDNWithResidualLoss_26182120636839
// MI455X (gfx1250) — compile-verified
//
#include <hip/hip_runtime.h>
#include <hip/hip_bf16.h>
#include <stdint.h>

// Problem constants (match reference)
#define BB    8
#define CC    81      // D+1 channels
#define HH    96
#define WW    320
#define NPERB 16      // boxes per image
#define CHUNK 27      // channels per TDM tile (81 = 3*27)
#define NCHNK 3
#define NWAVE (WW/32) // 10 waves per block

typedef __attribute__((ext_vector_type(4))) unsigned int v4u;
typedef __attribute__((ext_vector_type(8))) int          v8i;
typedef __attribute__((ext_vector_type(4))) int          v4i;

#if defined(__HIP_DEVICE_COMPILE__) && __has_builtin(__builtin_amdgcn_tensor_load_to_lds)
#define HAVE_TDM 1
#else
#define HAVE_TDM 0
#endif

#if defined(__HIP_DEVICE_COMPILE__) && __has_builtin(__builtin_amdgcn_s_wait_tensorcnt)
#define WAIT_TENSORCNT(n) __builtin_amdgcn_s_wait_tensorcnt(n)
#else
#define WAIT_TENSORCNT(n) asm volatile("s_wait_tensorcnt %0" :: "i"(n) : "memory")
#endif

// Low 32 bits of a flat shared-space pointer are the LDS byte offset
__device__ __forceinline__ uint32_t lds_byte_off(const void* p) {
    return (uint32_t)(uintptr_t)p;
}

#if HAVE_TDM
// Issue one 2D TDM load: `rows` rows of `rowElems` fp32 elements, row stride
// `rowStrideElems` (in elements), global -> LDS. Tracked by TENSORcnt.
__device__ __forceinline__ void tdm_load_rows(uint32_t ldsAddr, const float* gptr,
                                              int rows, int rowElems, long rowStrideElems) {
    uint64_t ga = (uint64_t)(uintptr_t)gptr;
    v4u g0;
    g0.x = 1u;                                              // count=1 (valid), user mode
    g0.y = ldsAddr;                                         // lds_addr [63:32]
    g0.z = (uint32_t)(ga & 0xFFFFFFFFu);                    // global_addr lo
    g0.w = (uint32_t)((ga >> 32) & 0x1FFFFFFu) | (2u << 30);// global_addr hi | type=2 (image)
    v8i g1;
    g1[0] = (2 << 16);                                      // data_size=2 -> 4 bytes; no pad/iter/multicast
    g1[1] = (int)(((uint32_t)rowElems & 0xFFFFu) << 16);    // tensor_dim0[15:0]  @ bits 63:48
    g1[2] = (int)((((uint32_t)rowElems >> 16) & 0xFFFFu)    // tensor_dim0[31:16] @ bits 79:64
                  | (((uint32_t)rows & 0xFFFFu) << 16));    // tensor_dim1[15:0]  @ bits 95:80
    g1[3] = (int)((((uint32_t)rows >> 16) & 0xFFFFu)        // tensor_dim1[31:16] @ bits 111:96
                  | (((uint32_t)rowElems & 0xFFFFu) << 16));// tile_dim0          @ bits 127:112
    g1[4] = (int)((uint32_t)rows & 0xFFFFu);                // tile_dim1 (tile_dim2 = 0)
    g1[5] = (int)(uint32_t)(rowStrideElems & 0xFFFFFFFF);   // tensor_dim0_stride lo32
    g1[6] = (int)(uint32_t)((rowStrideElems >> 32) & 0xFFFF);// tensor_dim0_stride hi16 (dim1_stride=0)
    g1[7] = 0;
    v4i zz4 = {0, 0, 0, 0};                                 // D# groups 2/3 unused (2D tensor)
    v8i zz8 = {0, 0, 0, 0, 0, 0, 0, 0};
    __builtin_amdgcn_tensor_load_to_lds(g0, g1, zz4, zz4, zz8, /*cpol*/0);
}
#endif

__global__ __launch_bounds__(WW) void ddn_loss_kernel(
    const float* __restrict__ logits,     // [B, C, H, W]
    const float* __restrict__ resid,      // [B, C, H, W]
    const float* __restrict__ boxes,      // [B*NPERB, 4]
    const float* __restrict__ cdepth,     // [B*NPERB]
    float* __restrict__ partial)          // [B*H, 2]
{
    __shared__ float sbuf[2][CHUNK * WW];                 // double-buffered logit tiles (69 KB)
    __shared__ float sU1[NPERB], sV1[NPERB], sU2[NPERB], sV2[NPERB], sDep[NPERB];
    __shared__ float swsum[NWAVE][2];

    const int w = threadIdx.x;
    const int h = blockIdx.x;
    const int b = blockIdx.y;

    // Stage the 16 boxes of this image into LDS (pre-floored/ceiled).
    if (w < NPERB) {
        int gi = b * NPERB + w;
        sU1[w] = floorf(boxes[gi * 4 + 0]);
        sV1[w] = floorf(boxes[gi * 4 + 1]);
        sU2[w] = ceilf (boxes[gi * 4 + 2]);
        sV2[w] = ceilf (boxes[gi * 4 + 3]);
        sDep[w] = cdepth[gi];
    }

    const float* gbase = logits + ((size_t)b * CC * HH + (size_t)h) * WW; // channel 0, row h
    const long  chanStride = (long)HH * WW;                               // elements between channels

#if HAVE_TDM
    if (threadIdx.x == 0) {
        tdm_load_rows(lds_byte_off(&sbuf[0][0]), gbase, CHUNK, WW, chanStride);
    }
#endif
    __syncthreads();   // boxes visible to all waves

    // ---- per-pixel depth map from boxes (min over covering boxes) ----
    const float fw = (float)w, fh = (float)h;
    float dm = 60.0f;
    bool  fg = false;
    #pragma unroll
    for (int i = 0; i < NPERB; ++i) {
        bool cov = (fw >= sU1[i]) & (fw < sU2[i]) & (fh >= sV1[i]) & (fh < sV2[i]);
        if (cov) { dm = fminf(dm, sDep[i]); fg = true; }
    }

    // ---- analytic target bin + residual target ----
    const float bin_size = 2.0f * (60.0f - 0.001f) / (80.0f * 81.0f);
    float idxf = -0.5f + 0.5f * sqrtf(1.0f + 8.0f * (dm - 0.001f) / bin_size);
    int target;
    if (!(idxf >= 0.0f) || idxf > 80.0f) target = 80;
    else                                 target = (int)floorf(idxf);
    if (target > 80) target = 80;
    float tf = (float)target;
    float wdepth = (target < 80)
        ? ((tf + 0.5f) * (tf + 0.5f) * bin_size * 0.5f - bin_size * 0.125f + 0.001f)
        : 60.0f;
    float res_t = dm - wdepth;

    // ---- online softmax over 81 channels, streamed through LDS tiles ----
    float m = -3.4e38f, ssum = 0.0f, xt = 0.0f;
    for (int ck = 0; ck < NCHNK; ++ck) {
#if HAVE_TDM
        if (threadIdx.x == 0) {
            if (ck + 1 < NCHNK)
                tdm_load_rows(lds_byte_off(&sbuf[(ck + 1) & 1][0]),
                              gbase + (size_t)(ck + 1) * CHUNK * chanStride,
                              CHUNK, WW, chanStride);
            if (ck + 1 < NCHNK) WAIT_TENSORCNT(1);  // in-order: oldest tile done
            else                WAIT_TENSORCNT(0);
        }
#else
        // Fallback: cooperative load of this chunk into LDS.
        for (int c = threadIdx.x >> 5; c < CHUNK; c += NWAVE) {
            int lane = threadIdx.x & 31;
            for (int x = lane; x < WW; x += 32)
                sbuf[ck & 1][c * WW + x] = gbase[(size_t)(ck * CHUNK + c) * chanStride + x];
        }
#endif
        __syncthreads();                            // tile ready for all waves
        const float* buf = &sbuf[ck & 1][0];
        const int c0 = ck * CHUNK;
        #pragma unroll
        for (int c = 0; c < CHUNK; ++c) {
            float x = buf[c * WW + w];
            if (c0 + c == target) xt = x;
            if (x > m) { ssum *= __expf(m - x); m = x; }   // rare rescale (~H(81) times)
            ssum += __expf(x - m);
        }
        __syncthreads();                            // tile consumed; buffer reusable
    }

    // ---- focal losses ----
    float pt    = __expf(xt - m) / ssum + 1e-8f;
    float om    = 1.0f - pt;
    float focal = 0.25f * om * om;
    float lmap  = -focal * __logf(pt);
    float pres  = resid[(((size_t)b * CC + target) * HH + h) * WW + w]; // gather at target channel
    float lres  = focal * fabsf(pres - res_t);
    float wt    = fg ? 13.0f : 1.0f;
    float a0 = lmap * wt;
    float a1 = lres * wt;

    // ---- deterministic block reduction: wave32 shfl -> LDS across 10 waves ----
    #pragma unroll
    for (int off = 16; off > 0; off >>= 1) {
        a0 += __shfl_down(a0, off, 32);
        a1 += __shfl_down(a1, off, 32);
    }
    const int wave = threadIdx.x >> 5;
    if ((threadIdx.x & 31) == 0) { swsum[wave][0] = a0; swsum[wave][1] = a1; }
    __syncthreads();
    if (threadIdx.x == 0) {
        float t0 = 0.0f, t1 = 0.0f;
        #pragma unroll
        for (int i = 0; i < NWAVE; ++i) { t0 += swsum[i][0]; t1 += swsum[i][1]; }
        const int blk = b * HH + h;
        partial[2 * blk + 0] = t0;
        partial[2 * blk + 1] = t1;
    }
}

__global__ __launch_bounds__(256) void ddn_reduce_kernel(const float* __restrict__ partial,
                                                         float* __restrict__ out)
{
    __shared__ float s0[256], s1[256];
    float t0 = 0.0f, t1 = 0.0f;
    for (int i = threadIdx.x; i < BB * HH; i += 256) {
        t0 += partial[2 * i + 0];
        t1 += partial[2 * i + 1];
    }
    s0[threadIdx.x] = t0;
    s1[threadIdx.x] = t1;
    __syncthreads();
    for (int off = 128; off > 0; off >>= 1) {
        if (threadIdx.x < off) {
            s0[threadIdx.x] += s0[threadIdx.x + off];
            s1[threadIdx.x] += s1[threadIdx.x + off];
        }
        __syncthreads();
    }
    if (threadIdx.x == 0) {
        const float inv = 1.0f / (float)((size_t)BB * HH * WW);
        out[0] = s0[0] * inv;
        out[1] = s1[0] * inv;
    }
}

extern "C" void kernel_launch(void* const* d_in, const int* in_sizes, int n_in,
                              void* d_out, int out_size, void* d_ws, size_t ws_size,
                              hipStream_t stream) {
    const float* logits = (const float*)d_in[0];  // depth_logits   [8,81,96,320] f32
    const float* resid  = (const float*)d_in[1];  // depth_residuals[8,81,96,320] f32
    const float* boxes  = (const float*)d_in[2];  // gt_boxes2d [128,4] f32
    // d_in[3]: num_gt_per_img (constant 16 per image; layout hardcoded to NPERB)
    const float* cdep   = (const float*)d_in[4];  // gt_center_depth [128] f32
    float* out     = (float*)d_out;               // 2 scalars
    float* partial = (float*)d_ws;                // [B*H, 2] per-block partials (6 KB)

    dim3 grid(HH, BB);
    ddn_loss_kernel<<<grid, WW, 0, stream>>>(logits, resid, boxes, cdep, partial);
    ddn_reduce_kernel<<<1, 256, 0, stream>>>(partial, out);
}